// DiagonalLinear_3478923510572
// MI455X (gfx1250) — compile-verified
//
#include <hip/hip_runtime.h>

// out[b, s, c] = x[b, s, c] * diag[c]
//   x:    [4, 4096, 4096] fp32  (256 MiB, streamed, no reuse)
//   diag: [4096]          fp32  (16 KiB, heavy reuse -> registers / L2)
//
// Pure HBM-bandwidth-bound (AI = 0.125 flop/byte). Floor at 23.3 TB/s is
// ~23 us. Strategy: wave32-coalesced b128 NT loads/stores, register-cached
// diag, global_prefetch_b8 of the next row.

typedef __attribute__((ext_vector_type(4))) float v4f;

#define DSIZE 4096          // last-axis length (== diag length)
#define TPB   256           // 8 wave32 per block
#define VPT   4             // float4 chunks per thread: 256 * 4 * 4 = 4096 cols

__global__ __launch_bounds__(TPB)
void diag_scale_kernel(const float* __restrict__ x,
                       const float* __restrict__ diag,
                       float* __restrict__ out,
                       int nrows)
{
    const int tid = threadIdx.x;

    // Register-cache this thread's slice of diag. Coalesced v4 loads
    // (lane stride = 16 B); diag stays resident in WGP$/L2 across blocks.
    v4f d[VPT];
#pragma unroll
    for (int j = 0; j < VPT; ++j) {
        const int col = j * (TPB * 4) + tid * 4;
        d[j] = *(const v4f*)(diag + col);
    }

    // Grid-stride over the 16384 rows; one 16 KiB row per iteration.
    for (int row = blockIdx.x; row < nrows; row += gridDim.x) {
        const size_t base = (size_t)row * DSIZE;

        // Prefetch the next row this block will process: 256 lanes at
        // 64 B stride cover the full 16 KiB row (-> global_prefetch_b8).
        const int nrow = row + gridDim.x;
        if (nrow < nrows) {
            __builtin_prefetch(x + (size_t)nrow * DSIZE + (size_t)tid * 16, 0, 0);
        }

#pragma unroll
        for (int j = 0; j < VPT; ++j) {
            const int col = j * (TPB * 4) + tid * 4;
            // Non-temporal: 512 MiB stream > 192 MB L2; don't pollute the
            // cache that holds diag.
            v4f xv = __builtin_nontemporal_load((const v4f*)(x + base + col));
            v4f r;
            r.x = xv.x * d[j].x;
            r.y = xv.y * d[j].y;
            r.z = xv.z * d[j].z;
            r.w = xv.w * d[j].w;
            __builtin_nontemporal_store(r, (v4f*)(out + base + col));
        }
    }
}

extern "C" void kernel_launch(void* const* d_in, const int* in_sizes, int n_in,
                              void* d_out, int out_size, void* d_ws, size_t ws_size,
                              hipStream_t stream)
{
    const float* x    = (const float*)d_in[0];
    const float* diag = (const float*)d_in[1];
    float*       out  = (float*)d_out;

    const int nrows = in_sizes[0] / DSIZE;          // 4 * 4096 = 16384 rows
    int blocks = nrows < 4096 ? nrows : 4096;       // 4 rows per block

    diag_scale_kernel<<<dim3(blocks), dim3(TPB), 0, stream>>>(x, diag, out, nrows);
}